// LightCurveDmDtBinner_18794776887600
// MI455X (gfx1250) — compile-verified
//
#include <hip/hip_runtime.h>
#include <math.h>

#define B_ 64
#define S_ 1024
#define DTB 24
#define DMB 24
#define NBINS (DTB*DMB)          // 576
#define NT (S_/16)               // 64 tile rows
#define NTILES (NT*(NT+1)/2)     // 2080 upper-tri tiles
#define SLICES 8
#define TPB 256
#define WAVES (TPB/32)

typedef __attribute__((ext_vector_type(2))) float v2f;
typedef __attribute__((ext_vector_type(8))) float v8f;

// ---------------- Phase 1: per-batch weighted std (loc cancels in dm) -------
__global__ __launch_bounds__(TPB)
void lcdmdt_stats_kernel(const float* __restrict__ lc, const float* __restrict__ vm,
                         float* __restrict__ inv_scale) {
  __shared__ float red[TPB];
  __shared__ float red2[TPB];
  __shared__ float s_loc, s_wsum;
  int b = blockIdx.x;
  const float* mag = lc + (size_t)b * 3 * S_ + S_;
  const float* w   = vm + (size_t)b * S_;
  float sw = 0.f, smw = 0.f;
  for (int s = threadIdx.x; s < S_; s += TPB) { float ww = w[s]; sw += ww; smw += mag[s] * ww; }
  red[threadIdx.x] = sw; red2[threadIdx.x] = smw; __syncthreads();
  for (int off = TPB / 2; off > 0; off >>= 1) {
    if (threadIdx.x < off) { red[threadIdx.x] += red[threadIdx.x + off]; red2[threadIdx.x] += red2[threadIdx.x + off]; }
    __syncthreads();
  }
  if (threadIdx.x == 0) { s_wsum = red[0]; s_loc = red2[0] / red[0]; }
  __syncthreads();
  float loc = s_loc, wsum = s_wsum;
  float sv = 0.f;
  for (int s = threadIdx.x; s < S_; s += TPB) { float d = mag[s] - loc; sv += d * d * w[s]; }
  red[threadIdx.x] = sv; __syncthreads();
  for (int off = TPB / 2; off > 0; off >>= 1) {
    if (threadIdx.x < off) red[threadIdx.x] += red[threadIdx.x + off];
    __syncthreads();
  }
  if (threadIdx.x == 0) inv_scale[b] = 1.0f / sqrtf(red[0] / wsum);
}

// ---------------- Phase 2: WMMA pairwise tiles + LDS histogram --------------
// Per 16x16 tile, three V_WMMA_F32_16X16X4_F32 ops produce:
//   dt[m][n] = t[n] - t[m]     (A = (-t[m], 1), B = (1, t[n])^T)  -- exact
//   dm[m][n] = m[n] - m[m]     (same construction)                 -- exact
//   dw[m][n] = w[m] * w[n]     (rank-1 outer product of 0/1 masks) -- exact
// which removes all per-row LDS weight loads from the inner loop.
__global__ __launch_bounds__(TPB)
void lcdmdt_hist_kernel(const float* __restrict__ lc, const float* __restrict__ vm,
                        const float* __restrict__ dt_edges, const float* __restrict__ dm_edges,
                        const float* __restrict__ inv_scale_g, float* __restrict__ hist_g) {
  __shared__ float lt[S_];
  __shared__ float lm[S_];
  __shared__ float lw[S_];
  __shared__ float lhist[NBINS];
  int b = blockIdx.x, slice = blockIdx.y;
  const float* lcb = lc + (size_t)b * 3 * S_;
  for (int s = threadIdx.x; s < S_; s += TPB) {
    lt[s] = lcb[s];
    lm[s] = lcb[S_ + s];
    lw[s] = vm[(size_t)b * S_ + s];
  }
  for (int h = threadIdx.x; h < NBINS; h += TPB) lhist[h] = 0.f;

  // edges are wave-uniform -> scalar loads / SGPRs
  float dte[25], dme[25];
#pragma unroll
  for (int k = 0; k < 25; ++k) { dte[k] = dt_edges[k]; dme[k] = dm_edges[k]; }
  float inv_scale = inv_scale_g[b];
  __syncthreads();

  const int wave = threadIdx.x >> 5;
  const int lane = threadIdx.x & 31;
  const int l16  = lane & 15;
  const bool hi  = lane >= 16;

  for (int u = slice * WAVES + wave; u < NTILES; u += SLICES * WAVES) {
    // map linear upper-tri tile index u -> (ti, tj), ti <= tj
    int t = (int)((float)NT - sqrtf(fmaxf((float)(NT * NT) - 2.0f * (float)u, 0.f)));
    if (t < 0) t = 0;
    if (t > NT - 1) t = NT - 1;
    int base = t * NT - (t * (t - 1)) / 2;
    while (t > 0 && base > u) { --t; base = t * NT - (t * (t - 1)) / 2; }
    for (;;) { int nb = (t + 1) * NT - ((t + 1) * t) / 2; if (nb <= u) { ++t; base = nb; } else break; }
    const int ti = t, tj = t + (u - base);
    const int i0 = ti * 16, j0 = tj * 16;

    float trow = lt[i0 + l16], tcol = lt[j0 + l16];
    float mrow = lm[i0 + l16], mcol = lm[j0 + l16];
    float wrow = lw[i0 + l16], wcol = lw[j0 + l16];

    // A 16x4 layout: lanes 0-15 hold M=lane, v0=K0, v1=K1; lanes 16-31 = K2,K3 (zero)
    // B 4x16 layout: v0 = {K0 row lanes 0-15, K1 row lanes 16-31}; v1 = K2,K3 (zero)
    v2f a1, b1, a2, b2, a3, b3;
    a1.x = hi ? 0.f : -trow;  a1.y = hi ? 0.f : 1.0f;
    b1.x = hi ? tcol : 1.0f;  b1.y = 0.f;
    a2.x = hi ? 0.f : -mrow;  a2.y = hi ? 0.f : 1.0f;
    b2.x = hi ? mcol : 1.0f;  b2.y = 0.f;
    a3.x = hi ? 0.f : wrow;   a3.y = 0.f;
    b3.x = hi ? 0.f : wcol;   b3.y = 0.f;
    v8f z = {};
    v8f cdt = __builtin_amdgcn_wmma_f32_16x16x4_f32(false, a1, false, b1, (short)0, z, false, false);
    v8f cdm = __builtin_amdgcn_wmma_f32_16x16x4_f32(false, a2, false, b2, (short)0, z, false, false);
    v8f cdw = __builtin_amdgcn_wmma_f32_16x16x4_f32(false, a3, false, b3, (short)0, z, false, false);

    const int N  = l16;
    const int gj = j0 + N;
#pragma unroll
    for (int r = 0; r < 8; ++r) {
      const int M  = r + (hi ? 8 : 0);
      const int gi = i0 + M;
      const float dwv = cdw[r];
      if (gi < gj && dwv != 0.f) {
        const float dtv = cdt[r];
        const float dmv = fabsf(cdm[r]) * inv_scale;
        int c1 = 0, c2 = 0;
#pragma unroll
        for (int k = 0; k < 25; ++k) { c1 += (dtv >= dte[k]); c2 += (dmv >= dme[k]); }
        int dti = c1 - 1; dti = dti < 0 ? 0 : (dti > DTB - 1 ? DTB - 1 : dti);
        int dmi = c2 - 1; dmi = dmi < 0 ? 0 : (dmi > DMB - 1 ? DMB - 1 : dmi);
        atomicAdd(&lhist[dmi * DTB + dti], dwv);   // ds_add_f32
      }
    }
  }
  __syncthreads();
  float* outp = hist_g + ((size_t)b * SLICES + slice) * NBINS;
  for (int h = threadIdx.x; h < NBINS; h += TPB) outp[h] = lhist[h];  // no global atomics
}

// ---------------- Phase 3: reduce slices, column-normalize, emit ------------
__global__ __launch_bounds__(TPB)
void lcdmdt_finalize_kernel(const float* __restrict__ hist_g, float* __restrict__ out) {
  __shared__ float h[NBINS];
  __shared__ float cnt[DTB];
  int b = blockIdx.x;
  for (int t = threadIdx.x; t < NBINS; t += TPB) {
    float s = 0.f;
#pragma unroll
    for (int sl = 0; sl < SLICES; ++sl) s += hist_g[((size_t)b * SLICES + sl) * NBINS + t];
    h[t] = s;
  }
  __syncthreads();
  if (threadIdx.x < DTB) {
    float s = 0.f;
#pragma unroll
    for (int d = 0; d < DMB; ++d) s += h[d * DTB + threadIdx.x];
    cnt[threadIdx.x] = s;
  }
  __syncthreads();
  for (int t = threadIdx.x; t < NBINS; t += TPB) {
    const int dti = t % DTB;
    const float c = cnt[dti];
    const bool valid = c > 0.f;
    const float denom = valid ? c : 1.f;
    float* ob = out + (size_t)b * 2 * NBINS;
    ob[t]         = valid ? h[t] / denom : 0.f;          // dm_given_dt [b][0][dmi][dti]
    ob[NBINS + t] = valid ? sqrtf(1.f / denom) : 0.f;    // uncertainty [b][1][dmi][dti]
    out[(size_t)B_ * 2 * NBINS + (size_t)b * NBINS + t] = valid ? 1.f : 0.f;  // valid flags
  }
}

extern "C" void kernel_launch(void* const* d_in, const int* in_sizes, int n_in,
                              void* d_out, int out_size, void* d_ws, size_t ws_size,
                              hipStream_t stream) {
  const float* lc   = (const float*)d_in[0];  // (B,3,S) f32
  const float* vm   = (const float*)d_in[1];  // (B,S)   f32
  const float* dt_e = (const float*)d_in[2];  // 25 f32
  const float* dm_e = (const float*)d_in[3];  // 25 f32
  float* ws = (float*)d_ws;
  float* inv_scale = ws;          // 64 floats
  float* hist      = ws + 64;     // B*SLICES*576 floats (~1.18 MB)

  lcdmdt_stats_kernel<<<B_, TPB, 0, stream>>>(lc, vm, inv_scale);
  lcdmdt_hist_kernel<<<dim3(B_, SLICES), TPB, 0, stream>>>(lc, vm, dt_e, dm_e, inv_scale, hist);
  lcdmdt_finalize_kernel<<<B_, TPB, 0, stream>>>(hist, (float*)d_out);
}